// DropChannel_20675972563785
// MI455X (gfx1250) — compile-verified
//
#include <hip/hip_runtime.h>

// DropChannel (weighted channel dropout), MI455X / gfx1250.
// Phase 1: per-channel spatial sums via V_WMMA_F32_16X16X4_F32 (A = ones).
// Phase 2: per-row WRS top-M selection + alpha (tiny, one block per n).
// Phase 3: bandwidth-bound broadcast scale with nontemporal float4 traffic.

#define N_     32
#define HW_    4096
#define C_     1024
#define M_     512          // WRS_RATIO * C
#define HSPLIT 8            // HW split for phase-1 parallelism
#define PKEEP  0.9f

typedef __attribute__((ext_vector_type(2))) float v2f;
typedef __attribute__((ext_vector_type(4))) float v4f;
typedef __attribute__((ext_vector_type(8))) float v8f;

// ---------------------------------------------------------------------------
// Phase 1: partial channel sums.  Each wave handles 16 channels x (HW/HSPLIT)
// rows.  B-tile = 4 HW-rows x 16 channels of x; A = ones(16x4) so
// D[m][c] accumulates sum_k B[k][c].  A all-ones makes the result invariant
// to the K-row<->lane permutation of the B operand layout; only the N (lane ->
// channel) mapping matters.  EXEC stays all-ones through every WMMA.
// ---------------------------------------------------------------------------
__global__ __launch_bounds__(256) void wcd_score_wmma(
    const float* __restrict__ x, float* __restrict__ part) {
  const int n     = blockIdx.z;
  const int hseg  = blockIdx.y;
  const int wave  = threadIdx.x >> 5;
  const int lane  = threadIdx.x & 31;
  const int c0    = blockIdx.x * 128 + wave * 16;   // 8 waves * 16 channels
  const int col   = lane & 15;
  const int rbase = (lane >> 4) << 1;               // rows {0,1} vs {2,3}
  const int hw0   = hseg * (HW_ / HSPLIT);

  const float* xp = x + ((size_t)n * HW_ + hw0 + rbase) * C_ + c0 + col;

  v2f a; a.x = 1.0f; a.y = 1.0f;                    // A = ones
  v8f acc = {};                                     // C/D accumulator (f32)
  for (int i = 0; i < HW_ / HSPLIT; i += 4) {
    v2f b;
    b.x = xp[0];        // x[n][hw0+i+rbase  ][c0+col]
    b.y = xp[C_];       // x[n][hw0+i+rbase+1][c0+col]
    xp += (size_t)4 * C_;
    acc = __builtin_amdgcn_wmma_f32_16x16x4_f32(
        false, a, false, b, (short)0, acc, false, false);
  }
  // C/D layout: VGPR0, lanes 0..15 hold D[M=0][N=lane] == channel sum.
  if (lane < 16)
    part[((size_t)hseg * N_ + n) * C_ + c0 + lane] = acc[0];
}

// ---------------------------------------------------------------------------
// Phase 2: one block per n (1024 threads = one thread per channel).
// Reduce HSPLIT partials -> score; key = r^(1/score) = exp(log(r)/score);
// exact rank selection reproducing thr = sort(key)[C-M], sel = key >= thr
// (tie-safe); LDS tree reductions for alpha; emit maskf[n][c].
// ---------------------------------------------------------------------------
__global__ __launch_bounds__(1024) void wcd_select(
    const float* __restrict__ part, const float* __restrict__ r,
    const float* __restrict__ u, float* __restrict__ maskf) {
  const int n = blockIdx.x;
  const int c = threadIdx.x;

  __shared__ float skey[C_];
  __shared__ float ssum[C_];
  __shared__ float sthr;

  float sc = 0.0f;
  for (int h = 0; h < HSPLIT; ++h)
    sc += part[((size_t)h * N_ + n) * C_ + c];
  sc *= (1.0f / (float)HW_);                         // spatial mean

  const float rv  = r[n * C_ + c];
  const float key = __expf(__logf(rv) / sc);         // r^(1/score)

  skey[c] = key;
  ssum[c] = sc;
  __syncthreads();

  // rank of this key: count strictly-less and less-or-equal
  int cnt_less = 0, cnt_leq = 0;
  for (int j = 0; j < C_; ++j) {
    const float kj = skey[j];
    cnt_less += (kj <  key);
    cnt_leq  += (kj <= key);
  }
  // sorted-ascending position range [cnt_less, cnt_leq-1] contains C-M?
  if (cnt_less <= (C_ - M_) && cnt_leq > (C_ - M_)) sthr = key;
  __syncthreads();

  const float thr = sthr;
  const bool  sel = (key >= thr);

  // reduce: total score and selected score
  __syncthreads();
  skey[c] = sel ? sc : 0.0f;                         // reuse skey for sel-score
  __syncthreads();
  for (int s = C_ / 2; s > 0; s >>= 1) {
    if (c < s) { skey[c] += skey[c + s]; ssum[c] += ssum[c + s]; }
    __syncthreads();
  }
  const float alpha = ssum[0] / skey[0];

  const bool rng = (u[n * C_ + c] < PKEEP);
  maskf[n * C_ + c] = (sel && rng) ? alpha : 0.0f;
}

// ---------------------------------------------------------------------------
// Phase 3: out[n][hw][c] = x[n][hw][c] * maskf[n][c].  Pure streaming:
// nontemporal float4 for x/out (no reuse, keep L2 clean); maskf (128 KB)
// stays resident in cache.
// ---------------------------------------------------------------------------
__global__ __launch_bounds__(256) void wcd_scale(
    const v4f* __restrict__ x4, const v4f* __restrict__ m4,
    v4f* __restrict__ out4) {
  const size_t idx = (size_t)blockIdx.x * 256 + threadIdx.x; // over N*HW*C/4
  const int    c4  = (int)(idx & (C_ / 4 - 1));              // 0..255
  const size_t row = idx >> 8;                               // n*HW + hw
  const int    n   = (int)(row >> 12);                       // / HW_

  v4f xv = __builtin_nontemporal_load(&x4[idx]);
  v4f mv = m4[(size_t)n * (C_ / 4) + c4];
  v4f ov;
  ov.x = xv.x * mv.x;
  ov.y = xv.y * mv.y;
  ov.z = xv.z * mv.z;
  ov.w = xv.w * mv.w;
  __builtin_nontemporal_store(ov, &out4[idx]);
}

// ---------------------------------------------------------------------------
extern "C" void kernel_launch(void* const* d_in, const int* in_sizes, int n_in,
                              void* d_out, int out_size, void* d_ws,
                              size_t ws_size, hipStream_t stream) {
  (void)in_sizes; (void)n_in; (void)out_size; (void)ws_size;
  const float* x = (const float*)d_in[0];   // (N, HW, C)
  const float* r = (const float*)d_in[1];   // (N, C)
  const float* u = (const float*)d_in[2];   // (N, C)
  float* out = (float*)d_out;

  float* part  = (float*)d_ws;                       // HSPLIT*N*C floats (1 MiB)
  float* maskf = part + (size_t)HSPLIT * N_ * C_;    // N*C floats (128 KiB)

  // Phase 1: grid (C/128 channel groups, HW segments, N)
  dim3 g1(C_ / 128, HSPLIT, N_);
  wcd_score_wmma<<<g1, 256, 0, stream>>>(x, part);

  // Phase 2: one block per image row n
  wcd_select<<<N_, C_, 0, stream>>>(part, r, u, maskf);

  // Phase 3: elementwise scale, N*HW*C/4 float4 lanes
  const size_t nvec = (size_t)N_ * HW_ * C_ / 4;     // 33,554,432
  wcd_scale<<<(unsigned)(nvec / 256), 256, 0, stream>>>(
      (const v4f*)x, (const v4f*)maskf, (v4f*)out);
}